// StructureLoss_43516608643714
// MI455X (gfx1250) — compile-verified
//
#include <hip/hip_runtime.h>

// CDNA5 WMMA fragment types: f32 16x16x4 -> A,B = 2 VGPRs (float2), C/D = 8 VGPRs (float8)
typedef float v2f __attribute__((ext_vector_type(2)));
typedef float v8f __attribute__((ext_vector_type(8)));

#define CC        64
#define FP_PITCH  129   // padded image row pitch (odd -> no LDS bank conflicts on column reads)
#define R_PITCH   65    // correlation accumulator pitch (odd -> conflict-free ds_add scatter)
#define SMEM_FLOATS (64*FP_PITCH + 2*64*R_PITCH)

// One workgroup per channel b. Computes, for input p in {x, target}:
//   R_p(dy,dx) = sum_{y,x} F[y,x] * F[y+dy, x+dx],   dy,dx in [-32,31]
// via WMMA GEMMs  C[y,(y',j)] = sum_x F[y,x] * Fp[y'][x+j]  (Fp = x-padded by 32),
// then scatters C onto R with ds_add_f32, then emits the channel's
// sum of ((Rx-Rt)/((64-|dy|)(64-|dx|)))^2 into partial[b].
__global__ __launch_bounds__(256)
void structloss_corr_kernel(const float* __restrict__ xin,
                            const float* __restrict__ tin,
                            float* __restrict__ partial)
{
    extern __shared__ float smem[];
    float* Fp = smem;                    // 64 x 129 padded image (one input at a time)
    float* R  = smem + 64*FP_PITCH;      // 2 x (64 x 65) correlation accumulators
    __shared__ float red[8];

    const int tid  = threadIdx.x;
    const int lane = tid & 31;
    const int wave = tid >> 5;
    const int b    = blockIdx.x;
    const int n16  = lane & 15;          // N index within fragment / M index for A
    const int hi   = lane >> 4;          // 0: lanes 0-15, 1: lanes 16-31

    // Static per-wave tile ownership: wave w -> y-tile (w>>1), y' half ((w&1)*32..+32)
    const int y0    = (wave >> 1) * 16;
    const int ypBeg = (wave & 1) * 32;

    for (int p = 0; p < 2; ++p) {
        __syncthreads();  // previous pass finished reading Fp / writing R[p-1]
        const float* src = (p == 0 ? xin : tin) + (size_t)b * (CC * CC);
        // Stage padded image: Fp[y][32..95] = F[y][0..63], zeros elsewhere
        for (int idx = tid; idx < 64 * 128; idx += 256) {
            int row = idx >> 7, col = idx & 127;
            Fp[row * FP_PITCH + col] =
                (col >= 32 && col < 96) ? src[row * 64 + (col - 32)] : 0.0f;
        }
        for (int idx = tid; idx < 64 * R_PITCH; idx += 256)
            R[p * 64 * R_PITCH + idx] = 0.0f;
        __syncthreads();

        // A fragments for this wave's y-tile, entire K=64 (16 chunks of 4), loaded once.
        // Layout (ISA 7.12.2, 32-bit A 16x4): lane m = lane%16, VGPR0=K(2*hi), VGPR1=K(2*hi+1)
        v2f afrag[16];
        {
            const float* arow = &Fp[(y0 + n16) * FP_PITCH + 32];
            #pragma unroll
            for (int kc = 0; kc < 16; ++kc) {
                int col = 4 * kc + 2 * hi;
                afrag[kc].x = arow[col];
                afrag[kc].y = arow[col + 1];
            }
        }

        float* Rp = &R[p * 64 * R_PITCH];
        for (int yp = ypBeg; yp < ypBeg + 32; ++yp) {
            int rel = yp - y0;                    // dy for M=0 is rel; tile dy range [rel-15, rel]
            if (rel < -32 || rel > 46) continue;  // whole tile outside dy in [-32,31] (uniform)
            const float* brow = &Fp[yp * FP_PITCH];
            for (int dxc = 0; dxc < 4; ++dxc) {
                const int j0 = dxc * 16;
                v8f acc = {0.f, 0.f, 0.f, 0.f, 0.f, 0.f, 0.f, 0.f};
                #pragma unroll
                for (int kc = 0; kc < 16; ++kc) {
                    // B layout (4x16): lane n = lane%16, VGPR0=K(2*hi), VGPR1=K(2*hi+1)
                    int base = 4 * kc + 2 * hi + j0 + n16;  // max 126 < FP_PITCH
                    v2f bfrag;
                    bfrag.x = brow[base];
                    bfrag.y = brow[base + 1];
                    acc = __builtin_amdgcn_wmma_f32_16x16x4_f32(
                        false, afrag[kc], false, bfrag,
                        (short)0, acc, false, false);
                }
                // Scatter: D element (M = v+8*hi, N = n16) -> R[yp - (y0+M) + 32][j0+n16]
                #pragma unroll
                for (int v = 0; v < 8; ++v) {
                    int M   = v + 8 * hi;
                    int dyi = rel - M + 32;
                    if (dyi >= 0 && dyi < 64)
                        atomicAdd(&Rp[dyi * R_PITCH + j0 + n16], acc[v]);
                }
            }
        }
    }
    __syncthreads();

    // Normalized squared difference, partial-summed over this channel's 64x64 lags
    float s = 0.f;
    for (int idx = tid; idx < 64 * 64; idx += 256) {
        int i = idx >> 6, j = idx & 63;
        int di = i - 32; if (di < 0) di = -di;
        int dj = j - 32; if (dj < 0) dj = -dj;
        float inv = 1.0f / ((float)(64 - di) * (float)(64 - dj));
        float d = (R[i * R_PITCH + j] - R[64 * R_PITCH + i * R_PITCH + j]) * inv;
        s += d * d;
    }
    for (int off = 16; off > 0; off >>= 1) s += __shfl_down(s, off, 32);
    if (lane == 0) red[wave] = s;
    __syncthreads();
    if (tid == 0) {
        float tot = 0.f;
        #pragma unroll
        for (int w = 0; w < 8; ++w) tot += red[w];
        partial[b] = tot;
    }
}

// Deterministic final reduction: mean over (1024 * 64 * 64) elements
__global__ __launch_bounds__(256)
void structloss_reduce_kernel(const float* __restrict__ partial,
                              float* __restrict__ out)
{
    __shared__ float red[8];
    const int tid = threadIdx.x, lane = tid & 31, wave = tid >> 5;
    float s = 0.f;
    for (int i = tid; i < 1024; i += 256) s += partial[i];
    for (int off = 16; off > 0; off >>= 1) s += __shfl_down(s, off, 32);
    if (lane == 0) red[wave] = s;
    __syncthreads();
    if (tid == 0) {
        float tot = 0.f;
        #pragma unroll
        for (int w = 0; w < 8; ++w) tot += red[w];
        out[0] = tot * (1.0f / 4194304.0f);  // / (1024 * 64 * 64)
    }
}

extern "C" void kernel_launch(void* const* d_in, const int* in_sizes, int n_in,
                              void* d_out, int out_size, void* d_ws, size_t ws_size,
                              hipStream_t stream) {
    const float* x = (const float*)d_in[0];
    const float* t = (const float*)d_in[1];
    float* partial = (float*)d_ws;        // 1024 floats of scratch
    float* out     = (float*)d_out;       // scalar loss

    const size_t smem_bytes = SMEM_FLOATS * sizeof(float);  // ~66 KB dynamic LDS
    structloss_corr_kernel<<<1024, 256, smem_bytes, stream>>>(x, t, partial);
    structloss_reduce_kernel<<<1, 256, 0, stream>>>(partial, out);
}